// URGCNLayer_71004399337807
// MI455X (gfx1250) — compile-verified
//
#include <hip/hip_runtime.h>
#include <hip/hip_bf16.h>

#define D 64
#define TILE 16

typedef __attribute__((ext_vector_type(2))) float v2f;
typedef __attribute__((ext_vector_type(8))) float v8f;

// ---------------------------------------------------------------------------
// Kernel 0: zero the workspace (S accumulator + counts)
// ---------------------------------------------------------------------------
__global__ void zero_ws_kernel(float* __restrict__ ws, int n) {
    int i = blockIdx.x * blockDim.x + threadIdx.x;
    int stride = gridDim.x * blockDim.x;
    for (; i < n; i += stride) ws[i] = 0.0f;
}

// ---------------------------------------------------------------------------
// Kernel 1: edge scatter.  64 threads per edge (one per feature dim).
//   S[dst][d]  += X[src][d] + R[rel][d]
//   cnt[dst]   += 1   (thread d==0)
// Gathers are coalesced per edge (contiguous 256B rows); everything lands in
// the 192MB L2 (S is 12.8MB), so atomics are L2-resident.
// ---------------------------------------------------------------------------
__global__ __launch_bounds__(256) void edge_scatter_kernel(
    const float* __restrict__ X, const float* __restrict__ R,
    const int* __restrict__ edges, float* __restrict__ S,
    float* __restrict__ cnt, int n_edges)
{
    int t = blockIdx.x * blockDim.x + threadIdx.x;
    int e = t >> 6;          // edge index (64 threads / edge)
    int d = t & 63;          // feature dim
    if (e >= n_edges) return;

    int src = edges[e * 3 + 0];
    int rel = edges[e * 3 + 1];
    int dst = edges[e * 3 + 2];

    float v = X[src * D + d] + R[rel * D + d];
    atomicAdd(&S[dst * D + d], v);
    if (d == 0) atomicAdd(&cnt[dst], 1.0f);
}

// ---------------------------------------------------------------------------
// Kernel 2: fused dual GEMM via V_WMMA_F32_16X16X4_F32:
//   out = X @ Ws  +  (S * rowscale) @ Wn ,  rowscale = 1/max(cnt,1)
// One wave (32 lanes) computes one 16x16 output tile. Block = 4 waves = the
// 4 column tiles of one 16-row stripe.
//
// A-operand (16x4 f32): lane holds row m = lane&15, K = k0 + (lane>>4)*2 .. +1
// B-operand (4x16 f32): lanes 0-15 hold K = k0..k0+1, lanes 16-31 K = k0+2..+3,
//                       column n = lane&15 (VGPR v -> K offset v within half)
// C/D (16x16 f32): VGPR r -> row r + (lane>=16 ? 8 : 0), col = lane&15
// ---------------------------------------------------------------------------
__global__ __launch_bounds__(128) void fused_gemm_kernel(
    const float* __restrict__ X, const float* __restrict__ Ws,
    const float* __restrict__ Wn, const float* __restrict__ S,
    const float* __restrict__ cnt, float* __restrict__ out, int n_nodes)
{
    const int lane  = threadIdx.x & 31;
    const int wave  = threadIdx.x >> 5;          // 0..3 -> N tile
    const int row0  = blockIdx.x * TILE;         // 16-row stripe
    const int n0    = wave * TILE;               // 16-col tile
    const int m     = lane & 15;                 // A row within tile
    const int khalf = (lane >> 4) * 2;           // A/B K sub-offset per lane half
    const int bn    = n0 + m;                    // B column for this lane

    if (row0 >= n_nodes) return;
    int arow = row0 + m;
    if (arow >= n_nodes) arow = n_nodes - 1;     // clamp (N_NODES%16==0 anyway)

    v8f acc = {};

    // ---- pass 1: X @ Ws ----
    const float* xrow = X + (size_t)arow * D;
    #pragma unroll
    for (int k0 = 0; k0 < D; k0 += 4) {
        v2f a = *(const v2f*)(xrow + k0 + khalf);
        v2f b;
        b[0] = Ws[(k0 + khalf)     * D + bn];
        b[1] = Ws[(k0 + khalf + 1) * D + bn];
        acc = __builtin_amdgcn_wmma_f32_16x16x4_f32(
            false, a, false, b, (short)0, acc, false, false);
    }

    // ---- pass 2: (S * 1/max(cnt,1)) @ Wn, accumulated into same tile ----
    const float inv = 1.0f / fmaxf(cnt[arow], 1.0f);
    const float* srow = S + (size_t)arow * D;
    #pragma unroll
    for (int k0 = 0; k0 < D; k0 += 4) {
        v2f a = *(const v2f*)(srow + k0 + khalf);
        a[0] *= inv;
        a[1] *= inv;
        v2f b;
        b[0] = Wn[(k0 + khalf)     * D + bn];
        b[1] = Wn[(k0 + khalf + 1) * D + bn];
        acc = __builtin_amdgcn_wmma_f32_16x16x4_f32(
            false, a, false, b, (short)0, acc, false, false);
    }

    // ---- store C tile ----
    const int mbase = row0 + ((lane >> 4) * 8);
    const int col   = n0 + (lane & 15);
    #pragma unroll
    for (int r = 0; r < 8; ++r) {
        int rr = mbase + r;
        if (rr < n_nodes) out[(size_t)rr * D + col] = acc[r];
    }
}

// ---------------------------------------------------------------------------
extern "C" void kernel_launch(void* const* d_in, const int* in_sizes, int n_in,
                              void* d_out, int out_size, void* d_ws, size_t ws_size,
                              hipStream_t stream)
{
    const float* nodes_embed    = (const float*)d_in[0];   // [N_NODES, 64]
    const float* relation_embed = (const float*)d_in[1];   // [N_REL, 64]
    const int*   edges          = (const int*)  d_in[2];   // [N_EDGES, 3]
    const float* w_self         = (const float*)d_in[3];   // [64, 64]
    const float* w_neighbor     = (const float*)d_in[4];   // [64, 64]
    float* out = (float*)d_out;

    const int n_nodes = in_sizes[0] / D;
    const int n_edges = in_sizes[2] / 3;

    float* S   = (float*)d_ws;            // [n_nodes, 64]
    float* cnt = S + (size_t)n_nodes * D; // [n_nodes]

    // 0) zero accumulators
    {
        int n = n_nodes * D + n_nodes;
        int blk = 256;
        int grid = (n + blk * 8 - 1) / (blk * 8);
        if (grid < 1) grid = 1;
        zero_ws_kernel<<<grid, blk, 0, stream>>>(S, n);
    }

    // 1) edge scatter (pre-GEMM aggregation: exploits linearity of @ Wn)
    {
        long long total = (long long)n_edges * 64;
        int blk = 256;
        int grid = (int)((total + blk - 1) / blk);
        edge_scatter_kernel<<<grid, blk, 0, stream>>>(
            nodes_embed, relation_embed, edges, S, cnt, n_edges);
    }

    // 2) fused WMMA GEMM: out = X@Ws + (S/cnt)@Wn
    {
        int grid = (n_nodes + TILE - 1) / TILE;   // 3125 stripes
        fused_gemm_kernel<<<grid, 128, 0, stream>>>(
            nodes_embed, w_self, w_neighbor, S, cnt, out, n_nodes);
    }
}